// LocalSpatioTemporalPooling_52355651338854
// MI455X (gfx1250) — compile-verified
//
#include <hip/hip_runtime.h>
#include <hip/hip_bf16.h>

// Shapes fixed by the reference:
//   x: (16, 256, 30, 64, 32) fp32  -> 983,040 contiguous 256-float stripe blocks
//   ff: (n, c, t, S) = block means, flat = 983,040 floats (lives in d_ws)
//   out: (16, 2048) fp32
#define N_DIM   16
#define C_DIM   256
#define T_DIM   30
#define S_DIM   8
#define NBLOCKS (N_DIM * C_DIM * T_DIM * S_DIM)   // 983040
#define COLS    (T_DIM * S_DIM)                   // 240 columns per n
#define FN      (C_DIM * COLS)                    // 61440 floats per n

typedef __attribute__((ext_vector_type(2))) float v2f;
typedef __attribute__((ext_vector_type(8))) float v8f;

// ---------------------------------------------------------------------------
// Phase 1: ff[b] = mean(x[b*256 .. b*256+256))  -- the HBM-bound 1 GB stream.
// One thread per block of 256 floats: 64 x b128 loads, float4 accumulate.
// Lanes are 1 KB apart; each 128B line is touched once and consumed over 8
// consecutive unrolled iterations out of L0, so DRAM traffic is exactly 1x.
// ---------------------------------------------------------------------------
__global__ void lstp_block_mean(const float* __restrict__ x,
                                float* __restrict__ ff) {
    const int tid = blockIdx.x * blockDim.x + threadIdx.x;   // < NBLOCKS
    const float4* __restrict__ p = reinterpret_cast<const float4*>(x) + (size_t)tid * 64;
    float ax = 0.f, ay = 0.f, az = 0.f, aw = 0.f;
#pragma unroll 8
    for (int i = 0; i < 64; ++i) {
        float4 v = p[i];
        ax += v.x; ay += v.y; az += v.z; aw += v.w;
    }
    ff[tid] = (ax + ay + az + aw) * (1.0f / 256.0f);
}

// ---------------------------------------------------------------------------
// Phase 2-4: one workgroup per n (512 threads = 16 waves).
//   Wave g (g<15) computes the 16x16 Gram tile of F's columns [16g,16g+16)
//   via V_WMMA_F32_16X16X4_F32 accumulating over c=256 in chunks of K=4,
//   with A == B (the 16x4 A layout and 4x16 B layout are lane-symmetric:
//   VGPR r holds K=r in lanes 0-15 and K=r+2 in lanes 16-31, M/N = lane&15).
//   Diagonal of the accumulator = squared column norms = scores.
// ---------------------------------------------------------------------------
__global__ void lstp_scores_topk(const float* __restrict__ ff,
                                 float* __restrict__ out) {
    __shared__ float stage[16][8][32];   // per-wave C/D dump for diag extract
    __shared__ float scores[COLS];       // 240 squared column norms (unclipped)
    __shared__ int   topi[S_DIM][2];

    const int n    = blockIdx.x;
    const int tid  = threadIdx.x;
    const int wave = tid >> 5;
    const int lane = tid & 31;
    const float* __restrict__ F = ff + (size_t)n * FN;   // row-major 256 x 240

    if (wave < 15) {
        const int col  = wave * 16 + (lane & 15);   // M == N == lane&15
        const int krow = (lane >> 4) * 2;           // K base: 0 or 2
        const float* __restrict__ p = F + (size_t)krow * COLS + col;
        v8f acc = {};
#pragma unroll 4
        for (int it = 0; it < 64; ++it) {           // c advances by 4 per WMMA
            v2f a;
            a.x = p[0];                             // K = krow     (VGPR0)
            a.y = p[COLS];                          // K = krow + 1 (VGPR1)
            p += 4 * COLS;
            acc = __builtin_amdgcn_wmma_f32_16x16x4_f32(
                false, a, false, a, (short)0, acc, false, false);
        }
#pragma unroll
        for (int r = 0; r < 8; ++r) stage[wave][r][lane] = acc[r];
    }
    __syncthreads();

    // Diagonal: D[m,m] -> m<8: VGPR m, lane m;  m>=8: VGPR m-8, lane m+16.
    if (wave < 15 && lane < 16) {
        const int m = lane;
        const float d = (m < 8) ? stage[wave][m][m] : stage[wave][m - 8][m + 16];
        scores[wave * 16 + m] = d;
    }
    __syncthreads();

    // Top-2 over t per stripe s; clip at EPS before compare (sqrt is monotone,
    // so comparing clipped sum-of-squares reproduces top_k on sqrt(clip(.))).
    // Strict '>' gives jax.lax.top_k's lowest-index tie-break.
    if (tid < S_DIM) {
        const int s = tid;
        const float EPS = 1e-6f;
        float b0 = -3.402823466e+38f; int i0 = 0;
        for (int t = 0; t < T_DIM; ++t) {
            float v = fmaxf(scores[t * S_DIM + s], EPS);
            if (v > b0) { b0 = v; i0 = t; }
        }
        float b1 = -3.402823466e+38f; int i1 = 0;
        for (int t = 0; t < T_DIM; ++t) {
            if (t == i0) continue;
            float v = fmaxf(scores[t * S_DIM + s], EPS);
            if (v > b1) { b1 = v; i1 = t; }
        }
        topi[s][0] = i0; topi[s][1] = i1;
    }
    __syncthreads();

    // out[n, s*256 + c] = 0.5 * (F[c, t0*8+s] + F[c, t1*8+s])
    for (int idx = tid; idx < S_DIM * C_DIM; idx += 512) {
        const int s = idx >> 8;
        const int c = idx & 255;
        const int t0 = topi[s][0], t1 = topi[s][1];
        const float v = 0.5f * (F[(size_t)c * COLS + t0 * S_DIM + s] +
                                F[(size_t)c * COLS + t1 * S_DIM + s]);
        out[(size_t)n * (S_DIM * C_DIM) + idx] = v;
    }
}

extern "C" void kernel_launch(void* const* d_in, const int* in_sizes, int n_in,
                              void* d_out, int out_size, void* d_ws, size_t ws_size,
                              hipStream_t stream) {
    const float* x  = (const float*)d_in[0];
    float*       out = (float*)d_out;
    float*       ff  = (float*)d_ws;   // 983,040 floats = 3.93 MB scratch

    // Phase 1: 983,040 block means, 256 threads/block -> 3840 workgroups.
    lstp_block_mean<<<NBLOCKS / 256, 256, 0, stream>>>(x, ff);
    // Phases 2-4: one workgroup per n, 16 waves (15 WMMA column-groups).
    lstp_scores_topk<<<N_DIM, 512, 0, stream>>>(ff, out);
}